// GraphSAGE_3049426780572
// MI455X (gfx1250) — compile-verified
//
#include <hip/hip_runtime.h>

typedef float v2f __attribute__((ext_vector_type(2)));
typedef float v8f __attribute__((ext_vector_type(8)));

#define HID 256

// ---------------------------------------------------------------------------
// Degree count: one thread per edge, f32 atomic add on deg[dst].
// ---------------------------------------------------------------------------
__global__ __launch_bounds__(256) void deg_kernel(const int* __restrict__ dst,
                                                  float* __restrict__ deg, int nE) {
    int e = blockIdx.x * 256 + threadIdx.x;
    if (e < nE) unsafeAtomicAdd(&deg[dst[e]], 1.0f);
}

__global__ __launch_bounds__(256) void invdeg_kernel(float* __restrict__ deg, int n) {
    int i = blockIdx.x * 256 + threadIdx.x;
    if (i < n) {
        float d = deg[i];
        deg[i] = (d > 0.0f) ? (1.0f / d) : 0.0f;
    }
}

// ---------------------------------------------------------------------------
// Scatter-sum: one wave32 per edge. Each lane moves float4 chunks of the
// source row and atomically adds into the destination row. Coalesced 512B
// reads per wave; f32 global atomics for the accumulate.
// ---------------------------------------------------------------------------
template <int C>
__global__ __launch_bounds__(256) void scatter_kernel(const float* __restrict__ h,
                                                      const int* __restrict__ src,
                                                      const int* __restrict__ dst,
                                                      float* __restrict__ agg, int nE) {
    int e = blockIdx.x * 8 + (threadIdx.x >> 5);
    if (e >= nE) return;
    int lane = threadIdx.x & 31;
    int s = src[e];
    int d = dst[e];
    const float4* hp = (const float4*)(h + (size_t)s * C);
    float* ap = agg + (size_t)d * C;
#pragma unroll
    for (int i = 0; i < C / 128; ++i) {
        float4 v = hp[lane + i * 32];
        int base = (lane + i * 32) * 4;
        unsafeAtomicAdd(ap + base + 0, v.x);
        unsafeAtomicAdd(ap + base + 1, v.y);
        unsafeAtomicAdd(ap + base + 2, v.z);
        unsafeAtomicAdd(ap + base + 3, v.w);
    }
}

// ---------------------------------------------------------------------------
// Fused SAGE layer GEMM:
//   out = relu( (agg * inv_deg) @ Wl + bl + h @ Wr )
// Block = 256 threads = 8 waves. Block owns 16 rows x 256 cols.
// Each wave: two 16x16 output tiles (cols wave*16 and wave*16+128),
// accumulated with V_WMMA_F32_16X16X4_F32 over K in steps of 4.
// A-tiles staged in LDS (inv_deg folded in), B streamed from L2-resident W.
//
// Fragment layouts (ISA 7.12.2):
//   A 16x4 f32 : a[v] = A[lane%16][k0 + v + 2*(lane/16)]
//   B 4x16 f32 : b[v] = B[k0 + v + 2*(lane/16)][n0 + lane%16]
//   C/D 16x16  : acc[v] -> row (v + 8*(lane/16)), col (lane%16)
// ---------------------------------------------------------------------------
template <int K>
__global__ __launch_bounds__(256) void sage_gemm_kernel(const float* __restrict__ agg,
                                                        const float* __restrict__ h,
                                                        const float* __restrict__ invdeg,
                                                        const float* __restrict__ Wl,
                                                        const float* __restrict__ bl,
                                                        const float* __restrict__ Wr,
                                                        float* __restrict__ out) {
    constexpr int LDK = K + 4;  // pad to dodge LDS bank conflicts
    __shared__ float sA[16 * LDK];
    __shared__ float sH[16 * LDK];

    const int m0 = blockIdx.x * 16;
    const int tid = threadIdx.x;

    // Cooperative stage of the 16xK agg tile (scaled by inv_deg) and h tile.
    for (int i = tid; i < 16 * (K / 4); i += 256) {
        int row = i / (K / 4);
        int kc = (i % (K / 4)) * 4;
        float s = invdeg[m0 + row];
        float4 va = *(const float4*)(agg + (size_t)(m0 + row) * K + kc);
        float4 vh = *(const float4*)(h + (size_t)(m0 + row) * K + kc);
        va.x *= s; va.y *= s; va.z *= s; va.w *= s;
        *(float4*)(&sA[row * LDK + kc]) = va;
        *(float4*)(&sH[row * LDK + kc]) = vh;
    }
    __syncthreads();

    const int wave = tid >> 5;
    const int lane = tid & 31;
    const int hf = lane >> 4;   // half-wave select
    const int l16 = lane & 15;
    const int n0 = wave * 16;
    const int n1 = n0 + 128;

    v8f acc0 = {};
    v8f acc1 = {};

#pragma unroll 4
    for (int k0 = 0; k0 < K; k0 += 4) {
        const int ka = k0 + 2 * hf;
        // agg @ Wl
        v2f a;
        a[0] = sA[l16 * LDK + ka];
        a[1] = sA[l16 * LDK + ka + 1];
        v2f b0, b1;
        b0[0] = Wl[(size_t)ka * HID + n0 + l16];
        b0[1] = Wl[(size_t)(ka + 1) * HID + n0 + l16];
        b1[0] = Wl[(size_t)ka * HID + n1 + l16];
        b1[1] = Wl[(size_t)(ka + 1) * HID + n1 + l16];
        acc0 = __builtin_amdgcn_wmma_f32_16x16x4_f32(false, a, false, b0, (short)0, acc0,
                                                     false, false);
        acc1 = __builtin_amdgcn_wmma_f32_16x16x4_f32(false, a, false, b1, (short)0, acc1,
                                                     false, false);
        // h @ Wr
        v2f c;
        c[0] = sH[l16 * LDK + ka];
        c[1] = sH[l16 * LDK + ka + 1];
        v2f d0, d1;
        d0[0] = Wr[(size_t)ka * HID + n0 + l16];
        d0[1] = Wr[(size_t)(ka + 1) * HID + n0 + l16];
        d1[0] = Wr[(size_t)ka * HID + n1 + l16];
        d1[1] = Wr[(size_t)(ka + 1) * HID + n1 + l16];
        acc0 = __builtin_amdgcn_wmma_f32_16x16x4_f32(false, c, false, d0, (short)0, acc0,
                                                     false, false);
        acc1 = __builtin_amdgcn_wmma_f32_16x16x4_f32(false, c, false, d1, (short)0, acc1,
                                                     false, false);
    }

    const float bias0 = bl[n0 + l16];
    const float bias1 = bl[n1 + l16];
#pragma unroll
    for (int v = 0; v < 8; ++v) {
        int row = m0 + v + 8 * hf;
        float o0 = acc0[v] + bias0;
        float o1 = acc1[v] + bias1;
        o0 = o0 > 0.0f ? o0 : 0.0f;
        o1 = o1 > 0.0f ? o1 : 0.0f;
        out[(size_t)row * HID + n0 + l16] = o0;
        out[(size_t)row * HID + n1 + l16] = o1;
    }
}

// ---------------------------------------------------------------------------
// Host-side orchestration (graph-capture safe: only async ops on `stream`).
// ---------------------------------------------------------------------------
extern "C" void kernel_launch(void* const* d_in, const int* in_sizes, int n_in,
                              void* d_out, int out_size, void* d_ws, size_t ws_size,
                              hipStream_t stream) {
    const float* x = (const float*)d_in[0];
    const int* edge = (const int*)d_in[1];
    const float* Wl0 = (const float*)d_in[2];
    const float* bl0 = (const float*)d_in[3];
    const float* Wr0 = (const float*)d_in[4];
    const float* Wl1 = (const float*)d_in[5];
    const float* bl1 = (const float*)d_in[6];
    const float* Wr1 = (const float*)d_in[7];
    const float* Wl2 = (const float*)d_in[8];
    const float* bl2 = (const float*)d_in[9];
    const float* Wr2 = (const float*)d_in[10];

    const int IN_C = 128;
    const int N = in_sizes[0] / IN_C;   // 50000
    const int E = in_sizes[1] / 2;      // 800000
    const int* src = edge;              // edge_index row 0
    const int* dst = edge + E;          // edge_index row 1

    float* ws = (float*)d_ws;
    float* invdeg = ws;                              // N floats
    float* agg = ws + (size_t)N;                     // N*HID floats
    float* h1 = agg + (size_t)N * HID;               // N*HID floats
    float* h2 = h1 + (size_t)N * HID;                // N*HID floats
    float* outp = (float*)d_out;

    const int egrid = (E + 255) / 256;
    const int ngrid = (N + 255) / 256;
    const int sgrid = (E + 7) / 8;      // one wave32 per edge, 8 waves/block
    const int ggrid = N / 16;           // 50000/16 = 3125 exactly

    // degree -> inverse degree
    hipMemsetAsync(invdeg, 0, (size_t)N * sizeof(float), stream);
    deg_kernel<<<egrid, 256, 0, stream>>>(dst, invdeg, E);
    invdeg_kernel<<<ngrid, 256, 0, stream>>>(invdeg, N);

    // ---- layer 0: x(128) -> h1(256)
    hipMemsetAsync(agg, 0, (size_t)N * IN_C * sizeof(float), stream);
    scatter_kernel<128><<<sgrid, 256, 0, stream>>>(x, src, dst, agg, E);
    sage_gemm_kernel<128><<<ggrid, 256, 0, stream>>>(agg, x, invdeg, Wl0, bl0, Wr0, h1);

    // ---- layer 1: h1(256) -> h2(256)
    hipMemsetAsync(agg, 0, (size_t)N * HID * sizeof(float), stream);
    scatter_kernel<256><<<sgrid, 256, 0, stream>>>(h1, src, dst, agg, E);
    sage_gemm_kernel<256><<<ggrid, 256, 0, stream>>>(agg, h1, invdeg, Wl1, bl1, Wr1, h2);

    // ---- layer 2: h2(256) -> out(256)
    hipMemsetAsync(agg, 0, (size_t)N * HID * sizeof(float), stream);
    scatter_kernel<256><<<sgrid, 256, 0, stream>>>(h2, src, dst, agg, E);
    sage_gemm_kernel<256><<<ggrid, 256, 0, stream>>>(agg, h2, invdeg, Wl2, bl2, Wr2, outp);
}